// AdaptLoss_directfuse_75591424409826
// MI455X (gfx1250) — compile-verified
//
#include <hip/hip_runtime.h>
#include <math.h>

// ---------------------------------------------------------------------------
// AdaptLoss (directfuse): mean |s_gt - (a*gt + b*t_gt)| with per-(n,c) gates
// from a tiny MLP over global-avg-pooled (gt, t_gt).
//
// Bandwidth-bound: ~251 MB of HBM traffic -> ~10.8us at 23.3 TB/s. Compute is
// negligible. We use V_WMMA_F32_16X16X4_F32 (B = ones) as a matrix-pipe
// cross-lane reducer (free vs VALU shuffles), and non-temporal loads for the
// read-once s_gt stream so the reused gt/t_gt (100 MB) stay resident in the
// 192 MB L2 between pass 1 and pass 2.
// ---------------------------------------------------------------------------

typedef __attribute__((ext_vector_type(2))) float v2f;
typedef __attribute__((ext_vector_type(4))) float f4;
typedef __attribute__((ext_vector_type(8))) float v8f;

#define HW          262144          // 512*512
#define NC          48              // N*C = 16*3
#define CCH         3
#define POOL_BLOCKS 32              // blocks per channel, pass 1
#define L1_BLOCKS   64              // blocks per channel, pass 2
#define INV_TOTAL   (1.0f / 12582912.0f)   // 1/(16*3*512*512)

// Wave32 all-lane sum via one f32 WMMA: D = A(16x4) * ones(4x16).
// A layout: lane m supplies K={0,1}, lane m+16 supplies K={2,3} for row m.
// With a = {x, 0}: rowsum[m] = x[m] + x[m+16]. Each lane's 8 C/D VGPRs hold
// 8 of the 16 rowsums (lanes 0-15 -> rows 0-7, lanes 16-31 -> rows 8-15), so
// sum(c[0..7]) gives a half-wave total; one xor-16 shuffle completes it.
// Requires EXEC == all ones (call only from uniform full-wave control flow).
__device__ __forceinline__ float wave_reduce_wmma(float x) {
  v2f a; a.x = x;   a.y = 0.0f;
  v2f b; b.x = 1.0f; b.y = 1.0f;
  v8f c = {};
  c = __builtin_amdgcn_wmma_f32_16x16x4_f32(false, a, false, b,
                                            (short)0, c, false, false);
  float s = c[0] + c[1] + c[2] + c[3] + c[4] + c[5] + c[6] + c[7];
  s += __shfl_xor(s, 16, 32);
  return s;
}

// Block (256 thr = 8 waves) reduction: WMMA wave reduce + tiny LDS combine.
__device__ __forceinline__ float block_reduce(float x, float* lds8) {
  x = wave_reduce_wmma(x);
  const int wave = threadIdx.x >> 5;
  const int lane = threadIdx.x & 31;
  if (lane == 0) lds8[wave] = x;
  __syncthreads();
  float tot = 0.0f;
  if (threadIdx.x == 0) {
    #pragma unroll
    for (int w = 0; w < 8; ++w) tot += lds8[w];
  }
  return tot;  // valid in thread 0 only
}

// ---- Pass 1: per-channel sums of gt and t_gt ------------------------------
// grid = (POOL_BLOCKS, NC), block = 256. Each block: 8192 floats per tensor.
__global__ void pool_kernel(const float* __restrict__ gt,
                            const float* __restrict__ tgt,
                            float* __restrict__ partials) {
  __shared__ float lds[2][8];
  const int nc  = blockIdx.y;
  const int blk = blockIdx.x;
  const size_t base = (size_t)nc * HW + (size_t)blk * (HW / POOL_BLOCKS);
  const f4* g4 = (const f4*)(gt  + base);
  const f4* t4 = (const f4*)(tgt + base);

  float sg = 0.0f, st = 0.0f;
  // 8192 floats = 2048 f4; 256 threads -> 8 vec4 loads each per tensor.
  for (int i = threadIdx.x; i < (HW / POOL_BLOCKS) / 4; i += 256) {
    f4 a = g4[i];
    f4 b = t4[i];
    sg += (a.x + a.y) + (a.z + a.w);
    st += (b.x + b.y) + (b.z + b.w);
  }
  sg = wave_reduce_wmma(sg);
  st = wave_reduce_wmma(st);
  const int wave = threadIdx.x >> 5, lane = threadIdx.x & 31;
  if (lane == 0) { lds[0][wave] = sg; lds[1][wave] = st; }
  __syncthreads();
  if (threadIdx.x == 0) {
    float a = 0.0f, b = 0.0f;
    #pragma unroll
    for (int w = 0; w < 8; ++w) { a += lds[0][w]; b += lds[1][w]; }
    partials[(0 * NC + nc) * POOL_BLOCKS + blk] = a;
    partials[(1 * NC + nc) * POOL_BLOCKS + blk] = b;
  }
}

// ---- Gate: reduce pool partials, run 2->32->2 MLP + softmax per (n,c) -----
// one block of 128 threads; microscopic work.
__global__ void gate_kernel(const float* __restrict__ partials,
                            const float* __restrict__ convW,  // [3][32][2]
                            const float* __restrict__ convB,  // [3][32]
                            const float* __restrict__ linW,   // [3][2][32]
                            const float* __restrict__ linB,   // [3][2]
                            float* __restrict__ ab) {         // [48][2]
  __shared__ float sums[2 * NC];
  const int t = threadIdx.x;
  if (t < 2 * NC) {
    float s = 0.0f;
    #pragma unroll
    for (int i = 0; i < POOL_BLOCKS; ++i) s += partials[t * POOL_BLOCKS + i];
    sums[t] = s;
  }
  __syncthreads();
  if (t < NC) {
    const int c = t % CCH;
    const float p0 = sums[t]      * (1.0f / HW);  // pooled gt
    const float p1 = sums[NC + t] * (1.0f / HW);  // pooled t_gt
    float l0 = linB[c * 2 + 0];
    float l1 = linB[c * 2 + 1];
    #pragma unroll
    for (int o = 0; o < 32; ++o) {
      const float f = fmaf(p0, convW[(c * 32 + o) * 2 + 0],
                      fmaf(p1, convW[(c * 32 + o) * 2 + 1], convB[c * 32 + o]));
      l0 = fmaf(f, linW[(c * 2 + 0) * 32 + o], l0);
      l1 = fmaf(f, linW[(c * 2 + 1) * 32 + o], l1);
    }
    const float m  = fmaxf(l0, l1);
    const float e0 = expf(l0 - m);
    const float e1 = expf(l1 - m);
    const float inv = 1.0f / (e0 + e1);
    ab[t * 2 + 0] = e0 * inv;
    ab[t * 2 + 1] = e1 * inv;
  }
}

// ---- Pass 2: sum |s_gt - (a*gt + b*t_gt)| ---------------------------------
// grid = (L1_BLOCKS, NC), block = 256. s_gt loaded non-temporally (read-once);
// gt/t_gt via regular loads (L2-resident from pass 1).
__global__ void l1_kernel(const float* __restrict__ gt,
                          const float* __restrict__ sgt,
                          const float* __restrict__ tgt,
                          const float* __restrict__ ab,
                          float* __restrict__ partials) {
  __shared__ float lds8[8];
  const int nc  = blockIdx.y;
  const int blk = blockIdx.x;
  const size_t base = (size_t)nc * HW + (size_t)blk * (HW / L1_BLOCKS);
  const float a = ab[nc * 2 + 0];
  const float b = ab[nc * 2 + 1];
  const f4* g4 = (const f4*)(gt  + base);
  const f4* s4 = (const f4*)(sgt + base);
  const f4* t4 = (const f4*)(tgt + base);

  float acc = 0.0f;
  // 4096 floats = 1024 f4; 256 threads -> 4 vec4 loads each per tensor.
  for (int i = threadIdx.x; i < (HW / L1_BLOCKS) / 4; i += 256) {
    f4 g  = g4[i];
    f4 s  = __builtin_nontemporal_load(&s4[i]);  // streamed, bypass reuse
    f4 tt = t4[i];
    acc += fabsf(s.x - fmaf(a, g.x, b * tt.x));
    acc += fabsf(s.y - fmaf(a, g.y, b * tt.y));
    acc += fabsf(s.z - fmaf(a, g.z, b * tt.z));
    acc += fabsf(s.w - fmaf(a, g.w, b * tt.w));
  }
  const float tot = block_reduce(acc, lds8);
  if (threadIdx.x == 0) partials[nc * L1_BLOCKS + blk] = tot;
}

// ---- Final deterministic reduction of 48*64 partials ----------------------
__global__ void final_kernel(const float* __restrict__ partials,
                             float* __restrict__ out) {
  __shared__ float lds8[8];
  float s = 0.0f;
  for (int i = threadIdx.x; i < NC * L1_BLOCKS; i += 256) s += partials[i];
  const float tot = block_reduce(s, lds8);
  if (threadIdx.x == 0) out[0] = tot * INV_TOTAL;  // LOSS_WEIGHT == 1
}

extern "C" void kernel_launch(void* const* d_in, const int* in_sizes, int n_in,
                              void* d_out, int out_size, void* d_ws, size_t ws_size,
                              hipStream_t stream) {
  const float* gt    = (const float*)d_in[0];
  const float* sgt   = (const float*)d_in[1];
  const float* tgt   = (const float*)d_in[2];
  const float* convW = (const float*)d_in[3];
  const float* convB = (const float*)d_in[4];
  const float* linW  = (const float*)d_in[5];
  const float* linB  = (const float*)d_in[6];

  float* W        = (float*)d_ws;
  float* poolPart = W;                  // 2*48*32 = 3072 floats
  float* ab       = W + 3072;           // 48*2    =   96 floats
  float* l1Part   = W + 3072 + 96;      // 48*64   = 3072 floats (~25 KB total)

  pool_kernel <<<dim3(POOL_BLOCKS, NC), 256, 0, stream>>>(gt, tgt, poolPart);
  gate_kernel <<<1, 128, 0, stream>>>(poolPart, convW, convB, linW, linB, ab);
  l1_kernel   <<<dim3(L1_BLOCKS, NC), 256, 0, stream>>>(gt, sgt, tgt, ab, l1Part);
  final_kernel<<<1, 256, 0, stream>>>(l1Part, (float*)d_out);
}